// FOPooling_89721866814194
// MI455X (gfx1250) — compile-verified
//
#include <hip/hip_runtime.h>
#include <hip/hip_bf16.h>
#include <stdint.h>

// ---------------------------------------------------------------------------
// QRNN fo-pooling, MI455X (gfx1250, wave32).
//   c_t = sigmoid(F)·c_{t-1} + (1-sigmoid(F))·tanh(Z);  h_t = sigmoid(O)·c_t
// Memory-bound (0.025 flop/byte): 3-phase chunked scan.
// Pass 3 streams each 48KB timestep slab through LDS with the gfx1250 Tensor
// Data Mover (4-deep ring, 3 slabs in flight), transcendentals use V_TANH_F32.
// ---------------------------------------------------------------------------

#define T_DIM   4096
#define B_DIM   16
#define H_DIM   256
#define H3      768
#define NCH     (B_DIM * H_DIM)        // 4096 channels
#define NCH4    (NCH / 4)              // 1024 float4 channel-groups
#define CHUNKS  256
#define CLEN    (T_DIM / CHUNKS)       // 16 timesteps / chunk
#define SLAB_ELTS  (B_DIM * H3)        // 12288 floats per timestep slab Y[t,:,:]
#define SLAB_BYTES (SLAB_ELTS * 4)     // 49152 bytes (contiguous)
#define DEPTH   4                      // TDM ring buffers (192KB of 320KB WGP LDS)

typedef float f32x4 __attribute__((ext_vector_type(4)));   // native vector for NT stores

__device__ __forceinline__ void nt_store4(float* p, float a, float b, float c, float d) {
  f32x4 v = {a, b, c, d};
  __builtin_nontemporal_store(v, (f32x4*)p);
}

// ---- fast transcendentals: gfx1250 has V_TANH_F32 (TRANS, co-executes) -----
__device__ __forceinline__ float fast_tanh(float x) {
#if defined(__AMDGCN__) && __has_builtin(__builtin_amdgcn_tanhf)
  return __builtin_amdgcn_tanhf(x);
#elif defined(__AMDGCN__) && __has_builtin(__builtin_amdgcn_tanh_f32)
  return __builtin_amdgcn_tanh_f32(x);
#else
  const float ax = fabsf(x);
  const float e  = __expf(-2.0f * ax);        // in (0,1]; no overflow path
  const float t  = (1.0f - e) / (1.0f + e);
  return copysignf(t, x);
#endif
}
__device__ __forceinline__ float sigf(float x) {  // sigmoid via tanh: 1 TRANS
  return 0.5f + 0.5f * fast_tanh(0.5f * x);
}

// ---------------- Pass 1: per-chunk affine reduction (A,B) ------------------
__global__ __launch_bounds__(1024) void qrnn_pass1(const float* __restrict__ Y,
                                                   float4* __restrict__ Aout,
                                                   float4* __restrict__ Bout) {
  const int j   = blockIdx.x;            // chunk
  const int tid = threadIdx.x;           // 0..1023
  const int b   = tid >> 6;              // 0..15
  const int h4  = (tid & 63) << 2;       // 0..252 step 4
  const float* base = Y + (size_t)j * CLEN * SLAB_ELTS + (size_t)b * H3 + h4;

  float4 a  = make_float4(1.f, 1.f, 1.f, 1.f);
  float4 bb = make_float4(0.f, 0.f, 0.f, 0.f);
#pragma unroll 4
  for (int i = 0; i < CLEN; ++i) {
    const float4 F = *(const float4*)(base + (size_t)i * SLAB_ELTS);
    const float4 Z = *(const float4*)(base + (size_t)i * SLAB_ELTS + H_DIM);
#define STEP1(c_) { float f = sigf(F.c_); float zf = (1.0f - f) * fast_tanh(Z.c_); \
                    a.c_ *= f; bb.c_ = f * bb.c_ + zf; }
    STEP1(x) STEP1(y) STEP1(z) STEP1(w)
#undef STEP1
  }
  Aout[(size_t)j * NCH4 + tid] = a;
  Bout[(size_t)j * NCH4 + tid] = bb;
}

// ---------------- Pass 2: scan over chunks, emit start states + c_last ------
__global__ __launch_bounds__(1024) void qrnn_pass2(const float* __restrict__ init_c,
                                                   const float4* __restrict__ Ain,
                                                   const float4* __restrict__ Bin,
                                                   float4* __restrict__ Cstart,
                                                   float* __restrict__ clast) {
  const int n = threadIdx.x;             // 0..1023 channel-group
  float4 c = ((const float4*)init_c)[n];
  for (int j = 0; j < CHUNKS; ++j) {
    Cstart[(size_t)j * NCH4 + n] = c;
    const float4 A = Ain[(size_t)j * NCH4 + n];
    const float4 B = Bin[(size_t)j * NCH4 + n];
    c.x = A.x * c.x + B.x;
    c.y = A.y * c.y + B.y;
    c.z = A.z * c.z + B.z;
    c.w = A.w * c.w + B.w;
  }
  ((float4*)clast)[n] = c;
}

// ---------------- Pass 3: replay chunk, TDM-staged through LDS --------------
#if defined(__AMDGCN__) && __has_builtin(__builtin_amdgcn_tensor_load_to_lds) && \
    __has_builtin(__builtin_amdgcn_s_wait_tensorcnt)
#define QRNN_TDM 1
typedef unsigned int u32x4 __attribute__((ext_vector_type(4)));
typedef int          i32x4 __attribute__((ext_vector_type(4)));
typedef int          i32x8 __attribute__((ext_vector_type(8)));

// Issue one TDM load of a contiguous 48KB timestep slab into LDS.
// D# per CDNA5 ISA ch.8: group0 = {count=1, lds_addr, global_addr, type=2};
// group1 = {data_size=4B, tensor_dim0=tile_dim0=12288, tensor_dim1=tile_dim1=1}.
__device__ __forceinline__ void tdm_load_slab(unsigned lds_byte_off,
                                              unsigned long long gaddr) {
  u32x4 g0;
  g0[0] = 1u;                                               // count=1 (valid), user mode
  g0[1] = lds_byte_off;                                     // lds_addr (bytes)
  g0[2] = (unsigned)gaddr;                                  // global_addr[31:0]
  g0[3] = (unsigned)((gaddr >> 32) & 0x01FFFFFFu) | 0x80000000u; // [56:32] | type=2
  i32x8 g1;
  g1[0] = (int)(2u << 16);                                  // data_size = 4 bytes
  g1[1] = (int)((SLAB_ELTS & 0xFFFFu) << 16);               // tensor_dim0[15:0]
  g1[2] = (int)(((unsigned)SLAB_ELTS >> 16) | (1u << 16));  // tensor_dim0[31:16], tensor_dim1=1
  g1[3] = (int)((SLAB_ELTS & 0xFFFFu) << 16);               // tile_dim0 = 12288
  g1[4] = 1;                                                // tile_dim1 = 1, tile_dim2 = 0
  g1[5] = SLAB_ELTS;                                        // tensor_dim0_stride
  g1[6] = 0;
  g1[7] = 0;
  i32x4 gz = {0, 0, 0, 0};                                  // groups 2/3 unused (<=2D)
#if __clang_major__ >= 23
  i32x8 gz8 = {0, 0, 0, 0, 0, 0, 0, 0};
  __builtin_amdgcn_tensor_load_to_lds(g0, g1, gz, gz, gz8, 0);
#else
  __builtin_amdgcn_tensor_load_to_lds(g0, g1, gz, gz, 0);
#endif
}
#endif

__global__ __launch_bounds__(1024) void qrnn_pass3(const float* __restrict__ Y,
                                                   const float4* __restrict__ Cstart,
                                                   float* __restrict__ Hout) {
  const int j   = blockIdx.x;
  const int tid = threadIdx.x;
  const int b   = tid >> 6;
  const int h4  = (tid & 63) << 2;
  float4 c = Cstart[(size_t)j * NCH4 + tid];

#ifdef QRNN_TDM
  extern __shared__ float smem[];  // DEPTH x 12288 floats (192KB of 320KB WGP LDS)
  const unsigned long long ybase =
      (unsigned long long)(uintptr_t)Y + (unsigned long long)j * CLEN * SLAB_BYTES;
  const bool wave0 = (tid >> 5) == 0;     // wave32: first wave issues all TDM ops
  if (wave0) {                            // prologue: DEPTH-1 slabs in flight
#pragma unroll
    for (int p = 0; p < DEPTH - 1; ++p)
      tdm_load_slab((unsigned)p * SLAB_BYTES, ybase + (unsigned long long)p * SLAB_BYTES);
  }

  for (int i = 0; i < CLEN; ++i) {
    const int cur = i & (DEPTH - 1);
    if (wave0) {
      if (i + DEPTH - 1 < CLEN) {
        tdm_load_slab((unsigned)((i + DEPTH - 1) & (DEPTH - 1)) * SLAB_BYTES,
                      ybase + (unsigned long long)(i + DEPTH - 1) * SLAB_BYTES);
        __builtin_amdgcn_s_wait_tensorcnt(DEPTH - 1);  // in-order: slab i complete
      } else {
        __builtin_amdgcn_s_wait_tensorcnt(0);          // tail: drain
      }
    }
    __syncthreads();  // release consumers once wave0 confirmed slab i is in LDS

    const float* buf = smem + cur * SLAB_ELTS + b * H3 + h4;
    const float4 F = *(const float4*)(buf);
    const float4 Z = *(const float4*)(buf + H_DIM);
    const float4 O = *(const float4*)(buf + 2 * H_DIM);
    float4 h;
#define STEP3(c_) { float f = sigf(F.c_); float zf = (1.0f - f) * fast_tanh(Z.c_); \
                    float o = sigf(O.c_); c.c_ = f * c.c_ + zf; h.c_ = o * c.c_; }
    STEP3(x) STEP3(y) STEP3(z) STEP3(w)
#undef STEP3
    const int t = j * CLEN + i;
    // NT store: h_out is write-once; keep L2 resident for Y's F/Z region.
    nt_store4(Hout + ((size_t)t * B_DIM + b) * H_DIM + h4, h.x, h.y, h.z, h.w);
    __syncthreads();  // all consumed buf[cur] before wave0 refills it
  }
#else
  // Fallback: plain vectorized global loads (also what the host pass parses).
  const float* base = Y + (size_t)j * CLEN * SLAB_ELTS + (size_t)b * H3 + h4;
#pragma unroll 4
  for (int i = 0; i < CLEN; ++i) {
    const float4 F = *(const float4*)(base + (size_t)i * SLAB_ELTS);
    const float4 Z = *(const float4*)(base + (size_t)i * SLAB_ELTS + H_DIM);
    const float4 O = *(const float4*)(base + (size_t)i * SLAB_ELTS + 2 * H_DIM);
    float4 h;
#define STEP3(c_) { float f = sigf(F.c_); float zf = (1.0f - f) * fast_tanh(Z.c_); \
                    float o = sigf(O.c_); c.c_ = f * c.c_ + zf; h.c_ = o * c.c_; }
    STEP3(x) STEP3(y) STEP3(z) STEP3(w)
#undef STEP3
    const int t = j * CLEN + i;
    nt_store4(Hout + ((size_t)t * B_DIM + b) * H_DIM + h4, h.x, h.y, h.z, h.w);
  }
#endif
}

// ---------------------------------------------------------------------------
extern "C" void kernel_launch(void* const* d_in, const int* in_sizes, int n_in,
                              void* d_out, int out_size, void* d_ws, size_t ws_size,
                              hipStream_t stream) {
  const float* Y      = (const float*)d_in[0];   // [T, B, 3H] f32
  const float* init_c = (const float*)d_in[1];   // [B, H] f32
  float* out = (float*)d_out;                    // h_out [T,B,H] ++ c_last [B,H]

  // Workspace: A (4MB) | B (4MB) | Cstart (4MB) = 12MB of d_ws.
  float4* Aw = (float4*)d_ws;
  float4* Bw = Aw + (size_t)CHUNKS * NCH4;
  float4* Cs = Bw + (size_t)CHUNKS * NCH4;

  qrnn_pass1<<<CHUNKS, 1024, 0, stream>>>(Y, Aw, Bw);
  qrnn_pass2<<<1, 1024, 0, stream>>>(init_c, Aw, Bw, Cs, out + (size_t)T_DIM * NCH);
  qrnn_pass3<<<CHUNKS, 1024, DEPTH * SLAB_BYTES, stream>>>(Y, Cs, out);
}